// GATNet_37538014167337
// MI455X (gfx1250) — compile-verified
//
#include <hip/hip_runtime.h>
#include <hip/hip_bf16.h>

typedef __attribute__((ext_vector_type(2))) float v2f;
typedef __attribute__((ext_vector_type(4))) float v4f;
typedef __attribute__((ext_vector_type(8))) float v8f;

#define NEG_SLOPE 0.2f
#define GEMM_TM 8          // M tiles (waves) per block
#define BPAD 132           // padded K-stride for LDS B panel (128 + 4)

// ---------------------------------------------------------------------------
// WMMA f32 GEMM: C[M x Ncols] = A[M x K] * B[K x Ncols], K == 128.
// Block = 256 threads = 8 wave32s. grid.x = ceil(Mtiles/8), grid.y = Ntiles.
// All 8 waves share one 16-column B panel staged in LDS (col-major, padded),
// each wave computes its own 16x16 tile with V_WMMA_F32_16X16X4_F32 (K/4 ops).
// A frag (16x4 f32): lanes 0-15 hold K={0,1}, lanes 16-31 K={2,3} -> b64 load
// B frag (4x16 f32): lanes 0-15 rows K={0,1}, lanes 16-31 rows K={2,3}
// C/D (16x16 f32):  VGPR i -> row i (lanes 0-15) / row i+8 (lanes 16-31)
// ---------------------------------------------------------------------------
__global__ void wmma_gemm_f32(const float* __restrict__ A,
                              const float* __restrict__ B,
                              float* __restrict__ C,
                              int M, int K, int Ncols) {
  __shared__ float Bs[16 * BPAD];          // Bs[col*BPAD + k]

  const int tid = threadIdx.x;
  const int n0  = blockIdx.y << 4;

  // cooperative stage of B panel: 16 cols x K rows (K==128 -> 2048 floats)
  for (int i = tid; i < 16 * 128; i += 256) {
    int k = i >> 4, col = i & 15;
    Bs[col * BPAD + k] = B[(size_t)k * Ncols + n0 + col];
  }
  __syncthreads();

  const int wave = tid >> 5;
  const int lane = tid & 31;
  const int half = lane >> 4;
  const int l16  = lane & 15;
  const int mt   = blockIdx.x * GEMM_TM + wave;
  if ((mt << 4) + 16 > M) return;          // wave-uniform guard (EXEC all-1s)
  const int m0 = mt << 4;

  const float* Arow = A + (size_t)(m0 + l16) * K;
  const float* BsL  = &Bs[l16 * BPAD];

  v8f acc = {};
  for (int k0 = 0; k0 < K; k0 += 4) {
    const int ka = k0 + half * 2;
    v2f a = *(const v2f*)(Arow + ka);      // global_load_b64 (K-contiguous)
    v2f b = *(const v2f*)(BsL + ka);       // ds_load_b64 (8B aligned: BPAD,ka even)
    acc = __builtin_amdgcn_wmma_f32_16x16x4_f32(false, a, false, b,
                                                (short)0, acc, false, false);
  }
  const int row = m0 + half * 8;
  const int col = n0 + l16;
#pragma unroll
  for (int i = 0; i < 8; ++i)
    C[(size_t)(row + i) * Ncols + col] = acc[i];
}

// ---------------------------------------------------------------------------
// Per-(node,head) attention logits: e_src = <h, a_src>, e_dst = <h, a_dst>
// C is a multiple of 4 -> float4 loads.
// ---------------------------------------------------------------------------
__global__ void attn_logits(const float* __restrict__ H,
                            const float* __restrict__ a_src,
                            const float* __restrict__ a_dst,
                            float* __restrict__ e_src,
                            float* __restrict__ e_dst,
                            int N, int heads, int C) {
  int idx = blockIdx.x * blockDim.x + threadIdx.x;
  if (idx >= N * heads) return;
  int n = idx / heads, h = idx % heads;
  const v4f* hp = (const v4f*)(H + (size_t)n * heads * C + (size_t)h * C);
  const v4f* as = (const v4f*)(a_src + (size_t)h * C);
  const v4f* ad = (const v4f*)(a_dst + (size_t)h * C);
  float s = 0.f, d = 0.f;
  for (int c = 0; c < (C >> 2); ++c) {
    v4f v = hp[c], va = as[c], vd = ad[c];
    s = fmaf(v.x, va.x, fmaf(v.y, va.y, fmaf(v.z, va.z, fmaf(v.w, va.w, s))));
    d = fmaf(v.x, vd.x, fmaf(v.y, vd.y, fmaf(v.z, vd.z, fmaf(v.w, vd.w, d))));
  }
  e_src[idx] = s;
  e_dst[idx] = d;
}

// src/dst for edge id (implicit self-loops appended at [E, E+N))
__device__ inline void edge_sd(const int* __restrict__ ei, int E, int e,
                               int& s, int& d) {
  if (e < E) { s = ei[e]; d = ei[E + e]; }
  else       { s = e - E; d = e - E; }
}

// order-preserving float<->uint mapping for atomic max
__device__ inline unsigned f2ord(float f) {
  unsigned u = __float_as_uint(f);
  return (u & 0x80000000u) ? ~u : (u | 0x80000000u);
}
__device__ inline float ord2f(unsigned u) {
  return __uint_as_float((u & 0x80000000u) ? (u & 0x7FFFFFFFu) : ~u);
}

__device__ inline float leaky(float v) {
  return v > 0.f ? v : v * NEG_SLOPE;
}

// Pass A: segment max of leaky-relu edge scores (ordered-uint atomic max)
template <int HEADS>
__global__ void edge_max(const int* __restrict__ ei, int E, int Etot,
                         const float* __restrict__ e_src,
                         const float* __restrict__ e_dst,
                         unsigned* __restrict__ m) {
  int e = blockIdx.x * blockDim.x + threadIdx.x;
  if (e >= Etot) return;
  int s, d; edge_sd(ei, E, e, s, d);
#pragma unroll
  for (int h = 0; h < HEADS; ++h) {
    float v = leaky(e_src[s * HEADS + h] + e_dst[d * HEADS + h]);
    atomicMax(&m[d * HEADS + h], f2ord(v));
  }
}

// Pass B: segment sum of exp(e - m)
template <int HEADS>
__global__ void edge_sumexp(const int* __restrict__ ei, int E, int Etot,
                            const float* __restrict__ e_src,
                            const float* __restrict__ e_dst,
                            const unsigned* __restrict__ m,
                            float* __restrict__ denom) {
  int e = blockIdx.x * blockDim.x + threadIdx.x;
  if (e >= Etot) return;
  int s, d; edge_sd(ei, E, e, s, d);
#pragma unroll
  for (int h = 0; h < HEADS; ++h) {
    float v  = leaky(e_src[s * HEADS + h] + e_dst[d * HEADS + h]);
    float ex = __expf(v - ord2f(m[d * HEADS + h]));
    atomicAdd(&denom[d * HEADS + h], ex);
  }
}

// Pass C: alpha-weighted scatter-aggregate. One thread per (edge, 4 channels):
// one b128 gather + 4 f32 atomics; alpha computed once (4 chans share a head,
// C == 32). All atomics land in the 192MB L2 (agg fits entirely).
template <int HEADS, int C>
__global__ void edge_aggregate(const int* __restrict__ ei, int E, int Etot,
                               const float* __restrict__ e_src,
                               const float* __restrict__ e_dst,
                               const unsigned* __restrict__ m,
                               const float* __restrict__ denom,
                               const float* __restrict__ H,
                               float* __restrict__ agg) {
  const int HC  = HEADS * C;
  const int QC  = HC >> 2;                     // float4 groups per edge
  long long gid   = (long long)blockIdx.x * blockDim.x + threadIdx.x;
  long long total = (long long)Etot * QC;
  if (gid >= total) return;
  int e   = (int)(gid / QC);
  int q   = (int)(gid % QC);
  int hc4 = q << 2;
  int h   = hc4 / C;
  int s, d; edge_sd(ei, E, e, s, d);
  float v     = leaky(e_src[s * HEADS + h] + e_dst[d * HEADS + h]);
  float ex    = __expf(v - ord2f(m[d * HEADS + h]));
  float alpha = ex / denom[d * HEADS + h];
  v4f msg = *(const v4f*)(H + (size_t)s * HC + hc4);   // global_load_b128
  float* ap = agg + (size_t)d * HC + hc4;
  atomicAdd(ap + 0, msg.x * alpha);
  atomicAdd(ap + 1, msg.y * alpha);
  atomicAdd(ap + 2, msg.z * alpha);
  atomicAdd(ap + 3, msg.w * alpha);
}

// bias + ELU (layer-1 epilogue), float4
__global__ void bias_elu(const float* __restrict__ agg,
                         const float* __restrict__ b,
                         float* __restrict__ out, int N, int F) {
  int idx = blockIdx.x * blockDim.x + threadIdx.x;     // over N*F/4
  if (idx >= N * (F >> 2)) return;
  int c4 = (idx % (F >> 2)) << 2;
  v4f v  = *(const v4f*)(agg + (size_t)idx * 4);
  v4f bb = *(const v4f*)(b + c4);
  v4f r;
  r.x = (v.x + bb.x); r.y = (v.y + bb.y); r.z = (v.z + bb.z); r.w = (v.w + bb.w);
  r.x = r.x > 0.f ? r.x : (__expf(r.x) - 1.f);
  r.y = r.y > 0.f ? r.y : (__expf(r.y) - 1.f);
  r.z = r.z > 0.f ? r.z : (__expf(r.z) - 1.f);
  r.w = r.w > 0.f ? r.w : (__expf(r.w) - 1.f);
  *(v4f*)(out + (size_t)idx * 4) = r;
}

// bias only (layer-2 epilogue; mean over 1 head == identity), float4
__global__ void bias_add(const float* __restrict__ agg,
                         const float* __restrict__ b,
                         float* __restrict__ out, int N, int F) {
  int idx = blockIdx.x * blockDim.x + threadIdx.x;     // over N*F/4
  if (idx >= N * (F >> 2)) return;
  int c4 = (idx % (F >> 2)) << 2;
  v4f v  = *(const v4f*)(agg + (size_t)idx * 4);
  v4f bb = *(const v4f*)(b + c4);
  v4f r;
  r.x = v.x + bb.x; r.y = v.y + bb.y; r.z = v.z + bb.z; r.w = v.w + bb.w;
  *(v4f*)(out + (size_t)idx * 4) = r;
}

extern "C" void kernel_launch(void* const* d_in, const int* in_sizes, int n_in,
                              void* d_out, int out_size, void* d_ws, size_t ws_size,
                              hipStream_t stream) {
  const float* x      = (const float*)d_in[0];
  const int*   ei     = (const int*)  d_in[1];   // [2, E] int32
  const float* W1     = (const float*)d_in[2];
  const float* a_src1 = (const float*)d_in[3];
  const float* a_dst1 = (const float*)d_in[4];
  const float* b1     = (const float*)d_in[5];
  const float* W2     = (const float*)d_in[6];
  const float* a_src2 = (const float*)d_in[7];
  const float* a_dst2 = (const float*)d_in[8];
  const float* b2     = (const float*)d_in[9];
  float* out = (float*)d_out;

  const int N    = in_sizes[0] / 128;  // 50000 (divisible by 16)
  const int E    = in_sizes[1] / 2;    // 800000
  const int Etot = E + N;              // with self loops
  const int H1F  = 128, H2F = 32;

  // workspace layout (f32 words)
  float* ws = (float*)d_ws;
  size_t off = 0;
  float*    H1   = ws + off; off += (size_t)N * H1F;
  float*    agg1 = ws + off; off += (size_t)N * H1F;
  float*    H2   = ws + off; off += (size_t)N * H2F;
  float*    agg2 = ws + off; off += (size_t)N * H2F;
  float*    es1  = ws + off; off += (size_t)N * 4;
  float*    ed1  = ws + off; off += (size_t)N * 4;
  float*    den1 = ws + off; off += (size_t)N * 4;
  unsigned* m1   = (unsigned*)(ws + off); off += (size_t)N * 4;
  float*    es2  = ws + off; off += (size_t)N;
  float*    ed2  = ws + off; off += (size_t)N;
  float*    den2 = ws + off; off += (size_t)N;
  unsigned* m2   = (unsigned*)(ws + off); off += (size_t)N;

  const int BLK = 256;
  const int mblocks = ((N / 16) + GEMM_TM - 1) / GEMM_TM;   // 391

  // ---- Layer 1 ----
  wmma_gemm_f32<<<dim3(mblocks, H1F / 16), 256, 0, stream>>>(x, W1, H1, N, 128, H1F);
  attn_logits<<<(N * 4 + BLK - 1) / BLK, BLK, 0, stream>>>(H1, a_src1, a_dst1,
                                                           es1, ed1, N, 4, 32);
  hipMemsetAsync(m1,   0, (size_t)N * 4 * 4, stream);
  hipMemsetAsync(den1, 0, (size_t)N * 4 * 4, stream);
  hipMemsetAsync(agg1, 0, (size_t)N * H1F * 4, stream);
  edge_max<4>   <<<(Etot + BLK - 1) / BLK, BLK, 0, stream>>>(ei, E, Etot, es1, ed1, m1);
  edge_sumexp<4><<<(Etot + BLK - 1) / BLK, BLK, 0, stream>>>(ei, E, Etot, es1, ed1, m1, den1);
  {
    long long total  = (long long)Etot * (H1F / 4);
    long long blocks = (total + BLK - 1) / BLK;
    edge_aggregate<4, 32><<<(unsigned)blocks, BLK, 0, stream>>>(
        ei, E, Etot, es1, ed1, m1, den1, H1, agg1);
  }
  // h1act = elu(agg1 + b1), reuse H1 buffer (H1 no longer needed)
  bias_elu<<<(N * (H1F / 4) + BLK - 1) / BLK, BLK, 0, stream>>>(agg1, b1, H1, N, H1F);

  // ---- Layer 2 ----
  wmma_gemm_f32<<<dim3(mblocks, H2F / 16), 256, 0, stream>>>(H1, W2, H2, N, 128, H2F);
  attn_logits<<<(N + BLK - 1) / BLK, BLK, 0, stream>>>(H2, a_src2, a_dst2,
                                                       es2, ed2, N, 1, 32);
  hipMemsetAsync(m2,   0, (size_t)N * 4, stream);
  hipMemsetAsync(den2, 0, (size_t)N * 4, stream);
  hipMemsetAsync(agg2, 0, (size_t)N * H2F * 4, stream);
  edge_max<1>   <<<(Etot + BLK - 1) / BLK, BLK, 0, stream>>>(ei, E, Etot, es2, ed2, m2);
  edge_sumexp<1><<<(Etot + BLK - 1) / BLK, BLK, 0, stream>>>(ei, E, Etot, es2, ed2, m2, den2);
  {
    long long total  = (long long)Etot * (H2F / 4);
    long long blocks = (total + BLK - 1) / BLK;
    edge_aggregate<1, 32><<<(unsigned)blocks, BLK, 0, stream>>>(
        ei, E, Etot, es2, ed2, m2, den2, H2, agg2);
  }
  bias_add<<<(N * (H2F / 4) + BLK - 1) / BLK, BLK, 0, stream>>>(agg2, b2, out, N, H2F);
}